// VisionMambaBlock_87729001988424
// MI455X (gfx1250) — compile-verified
//
#include <hip/hip_runtime.h>

typedef float v2f __attribute__((ext_vector_type(2)));
typedef float v8f __attribute__((ext_vector_type(8)));

#define D_MODEL 192
#define D_STATE 16
#define D_INNER 384
#define DT_RANK 12
#define XDBL_COLS 48   // 44 padded to 48 (3 WMMA N-tiles)
#define L_SEQ 64
#define N_TOK 32768    // 8 * 64 * 64
#define HW 4096        // 64 * 64

__device__ __forceinline__ v8f wmma_f32(v2f a, v2f b, v8f c) {
    // D = A(16x4) * B(4x16) + C, full f32, RNE
    return __builtin_amdgcn_wmma_f32_16x16x4_f32(false, a, false, b, (short)0, c,
                                                 false, false);
}

// ---------------------------------------------------------------------------
// Kernel 1: xz[t, 0:768] = u[t, 0:192] @ W_in, with u gathered (transposed)
// from x[B,C,H,W]. Grid: (NT/16, 6), block 256 (8 waves -> N=128 per block).
// ---------------------------------------------------------------------------
__global__ void k_inproj(const float* __restrict__ x, const float* __restrict__ W_in,
                         float* __restrict__ xz) {
    __shared__ float Alds[D_MODEL * 16];       // [k][m], m = token-in-tile
    const int t0 = blockIdx.x * 16;            // 16 tokens, same (b,h), w0..w0+15
    const int b  = t0 >> 12;
    const int hh = (t0 >> 6) & 63;
    const int w0 = t0 & 63;
    const float* xbase = x + (size_t)b * D_MODEL * HW + hh * 64 + w0;

    for (int idx = threadIdx.x; idx < D_MODEL * 16; idx += blockDim.x) {
        const int c = idx >> 4, m = idx & 15;  // coalesced over w
        Alds[c * 16 + m] = xbase[(size_t)c * HW + m];
    }
    __syncthreads();

    const int wave = threadIdx.x >> 5;
    const int lane = threadIdx.x & 31;
    const int m    = lane & 15;
    const int kl   = (lane >> 4) * 2;
    const int nc   = blockIdx.y * 128 + wave * 16 + (lane & 15);

    v8f acc = {};
    for (int k = 0; k < D_MODEL; k += 4) {
        v2f a, bm;
        a.x  = Alds[(k + kl) * 16 + m];
        a.y  = Alds[(k + kl + 1) * 16 + m];
        bm.x = W_in[(size_t)(k + kl) * 768 + nc];
        bm.y = W_in[(size_t)(k + kl + 1) * 768 + nc];
        acc = wmma_f32(a, bm, acc);
    }
    const int mrow0 = (lane >> 4) * 8;
#pragma unroll
    for (int r = 0; r < 8; ++r)
        xz[(size_t)(t0 + mrow0 + r) * 768 + nc] = acc[r];
}

// ---------------------------------------------------------------------------
// Kernel 2: depthwise causal conv (D_CONV=4) + bias + SiLU.
// Grid: NT blocks, block = 384 threads (one d per thread, coalesced).
// ---------------------------------------------------------------------------
__global__ void k_conv(const float* __restrict__ xz, const float* __restrict__ conv_w,
                       const float* __restrict__ conv_b, float* __restrict__ xb) {
    const int t = blockIdx.x;
    const int d = threadIdx.x;
    const int w = t & 63;
    float s = conv_b[d];
#pragma unroll
    for (int j = 0; j < 4; ++j) {
        const int wj = w - 3 + j;
        if (wj >= 0) s += conv_w[d * 4 + j] * xz[(size_t)(t - 3 + j) * 768 + d];
    }
    xb[(size_t)t * D_INNER + d] = s / (1.0f + __expf(-s));  // SiLU
}

// ---------------------------------------------------------------------------
// Kernel 3: x_dbl[t, 0:44] = x[t, 0:384] @ W_x (N padded to 48).
// Grid: NT/16, block 128 (4 waves; waves 0..2 compute one N-tile each).
// ---------------------------------------------------------------------------
__global__ void k_xdbl(const float* __restrict__ xb, const float* __restrict__ W_x,
                       float* __restrict__ xdbl) {
    __shared__ float Alds[D_INNER * 17];       // [k][m], stride 17 kills conflicts
    const int t0 = blockIdx.x * 16;
    for (int idx = threadIdx.x; idx < D_INNER * 16; idx += blockDim.x) {
        const int mm = idx / D_INNER;
        const int kk = idx - mm * D_INNER;     // coalesced global read
        Alds[kk * 17 + mm] = xb[(size_t)(t0 + mm) * D_INNER + kk];
    }
    __syncthreads();

    const int wave = threadIdx.x >> 5;
    const int lane = threadIdx.x & 31;
    if (wave < 3) {
        const int m  = lane & 15;
        const int kl = (lane >> 4) * 2;
        const int nn = wave * 16 + (lane & 15);
        v8f acc = {};
        for (int k = 0; k < D_INNER; k += 4) {
            v2f a, bm;
            a.x  = Alds[(k + kl) * 17 + m];
            a.y  = Alds[(k + kl + 1) * 17 + m];
            bm.x = (nn < 44) ? W_x[(size_t)(k + kl) * 44 + nn] : 0.0f;
            bm.y = (nn < 44) ? W_x[(size_t)(k + kl + 1) * 44 + nn] : 0.0f;
            acc = wmma_f32(a, bm, acc);
        }
        const int mrow0 = (lane >> 4) * 8;
#pragma unroll
        for (int r = 0; r < 8; ++r)
            xdbl[(size_t)(t0 + mrow0 + r) * XDBL_COLS + nn] = acc[r];
    }
}

// ---------------------------------------------------------------------------
// Kernel 4: selective scan. One workgroup per sequence, 384 threads (12 waves).
// Thread d keeps h[16], -exp(A_log)[16], W_dt column[12] in registers.
// Fused: delta = softplus(dt@W_dt + b_dt); y = scan + x*D; y *= silu(z).
// Writes y in place over xb.
// ---------------------------------------------------------------------------
__global__ void k_scan(const float* __restrict__ xz, float* __restrict__ xb,
                       const float* __restrict__ xdbl, const float* __restrict__ W_dt,
                       const float* __restrict__ b_dt, const float* __restrict__ A_log,
                       const float* __restrict__ Dp) {
    __shared__ float row[XDBL_COLS];           // dt[0:12] | B[12:28] | C[28:44]
    const int s = blockIdx.x;
    const int d = threadIdx.x;

    float Ad[D_STATE];
#pragma unroll
    for (int n = 0; n < D_STATE; ++n) Ad[n] = -__expf(A_log[d * D_STATE + n]);
    float wdt[DT_RANK];
#pragma unroll
    for (int k = 0; k < DT_RANK; ++k) wdt[k] = W_dt[k * D_INNER + d];
    const float bd = b_dt[d];
    const float Dd = Dp[d];

    float h[D_STATE];
#pragma unroll
    for (int n = 0; n < D_STATE; ++n) h[n] = 0.0f;

    const int tbase = s * L_SEQ;
    for (int l = 0; l < L_SEQ; ++l) {
        const int t = tbase + l;
        if (threadIdx.x < XDBL_COLS)
            row[threadIdx.x] = xdbl[(size_t)t * XDBL_COLS + threadIdx.x];
        __syncthreads();

        float dtv = bd;
#pragma unroll
        for (int k = 0; k < DT_RANK; ++k) dtv += row[k] * wdt[k];
        const float delta = (dtv > 20.0f) ? dtv : __logf(1.0f + __expf(dtv));

        const float xv = xb[(size_t)t * D_INNER + d];
        const float dx = delta * xv;
        float y = 0.0f;
#pragma unroll
        for (int n = 0; n < D_STATE; ++n) {
            h[n] = __expf(delta * Ad[n]) * h[n] + dx * row[DT_RANK + n];
            y += h[n] * row[DT_RANK + D_STATE + n];
        }
        y += xv * Dd;
        const float zv = xz[(size_t)t * 768 + D_INNER + d];
        y *= zv / (1.0f + __expf(-zv));        // silu gate
        xb[(size_t)t * D_INNER + d] = y;       // in place
        __syncthreads();
    }
}

// ---------------------------------------------------------------------------
// Kernel 5: out = LayerNorm(y @ W_out) scattered back to [B,C,H,W].
// Grid: NT/16, block 384 (12 waves = 12 N-tiles of 16 = 192 channels).
// ---------------------------------------------------------------------------
__global__ void k_outln(const float* __restrict__ y, const float* __restrict__ W_out,
                        const float* __restrict__ ln_g, const float* __restrict__ ln_b,
                        float* __restrict__ out) {
    __shared__ float Alds[D_INNER * 17];
    __shared__ float yout[16 * 193];           // stride 193: channel reads conflict-free
    __shared__ float mu[16], rv[16];
    const int t0 = blockIdx.x * 16;

    for (int idx = threadIdx.x; idx < D_INNER * 16; idx += blockDim.x) {
        const int mm = idx / D_INNER;
        const int kk = idx - mm * D_INNER;
        Alds[kk * 17 + mm] = y[(size_t)(t0 + mm) * D_INNER + kk];
    }
    __syncthreads();

    const int wave = threadIdx.x >> 5;
    const int lane = threadIdx.x & 31;
    const int m    = lane & 15;
    const int kl   = (lane >> 4) * 2;
    const int nn   = wave * 16 + (lane & 15);

    v8f acc = {};
    for (int k = 0; k < D_INNER; k += 4) {
        v2f a, bm;
        a.x  = Alds[(k + kl) * 17 + m];
        a.y  = Alds[(k + kl + 1) * 17 + m];
        bm.x = W_out[(size_t)(k + kl) * D_MODEL + nn];
        bm.y = W_out[(size_t)(k + kl + 1) * D_MODEL + nn];
        acc = wmma_f32(a, bm, acc);
    }
    const int mrow0 = (lane >> 4) * 8;
#pragma unroll
    for (int r = 0; r < 8; ++r) yout[(mrow0 + r) * 193 + nn] = acc[r];
    __syncthreads();

    if (threadIdx.x < 16) {
        float sum = 0.0f, ssq = 0.0f;
        for (int c = 0; c < D_MODEL; ++c) {
            const float v = yout[threadIdx.x * 193 + c];
            sum += v; ssq += v * v;
        }
        const float mean = sum * (1.0f / D_MODEL);
        const float var  = ssq * (1.0f / D_MODEL) - mean * mean;  // biased (jnp.var)
        mu[threadIdx.x] = mean;
        rv[threadIdx.x] = rsqrtf(var + 1e-5f);
    }
    __syncthreads();

    const int b  = t0 >> 12;
    const int hh = (t0 >> 6) & 63;
    const int w0 = t0 & 63;
    float* ob = out + (size_t)b * D_MODEL * HW + hh * 64 + w0;
    for (int e = threadIdx.x; e < 16 * D_MODEL; e += blockDim.x) {
        const int c = e >> 4, mm = e & 15;     // coalesced over w
        const float v = (yout[mm * 193 + c] - mu[mm]) * rv[mm] * ln_g[c] + ln_b[c];
        ob[(size_t)c * HW + mm] = v;
    }
}

// ---------------------------------------------------------------------------
extern "C" void kernel_launch(void* const* d_in, const int* in_sizes, int n_in,
                              void* d_out, int out_size, void* d_ws, size_t ws_size,
                              hipStream_t stream) {
    const float* x      = (const float*)d_in[0];
    const float* W_in   = (const float*)d_in[1];
    const float* conv_w = (const float*)d_in[2];
    const float* conv_b = (const float*)d_in[3];
    const float* W_x    = (const float*)d_in[4];
    const float* W_dt   = (const float*)d_in[5];
    const float* b_dt   = (const float*)d_in[6];
    const float* A_log  = (const float*)d_in[7];
    const float* Dp     = (const float*)d_in[8];
    const float* W_out  = (const float*)d_in[9];
    const float* ln_g   = (const float*)d_in[10];
    const float* ln_b   = (const float*)d_in[11];
    (void)in_sizes; (void)n_in; (void)out_size; (void)ws_size;

    float* ws   = (float*)d_ws;
    float* xz   = ws;                                // NT*768  (100.7 MB)
    float* xb   = xz + (size_t)N_TOK * 768;          // NT*384  (x, then y in place)
    float* xdbl = xb + (size_t)N_TOK * D_INNER;      // NT*48
    float* out  = (float*)d_out;

    dim3 g1(N_TOK / 16, 6);
    k_inproj<<<g1, 256, 0, stream>>>(x, W_in, xz);
    k_conv  <<<N_TOK, D_INNER, 0, stream>>>(xz, conv_w, conv_b, xb);
    k_xdbl  <<<N_TOK / 16, 128, 0, stream>>>(xb, W_x, xdbl);
    k_scan  <<<512, D_INNER, 0, stream>>>(xz, xb, xdbl, W_dt, b_dt, A_log, Dp);
    k_outln <<<N_TOK / 16, 384, 0, stream>>>(xb, W_out, ln_g, ln_b, out);
}